// PGA_Unfold_J10_PC_22883585753567
// MI455X (gfx1250) — compile-verified
//
#include <hip/hip_runtime.h>
#include <math.h>

// ---------------- problem constants (from reference) ----------------
constexpr int NT   = 64;    // Nt antennas
constexpr int NRF  = 8;     // RF chains
constexpr int MU   = 4;     // users M
constexpr int KC   = 4;     // subcarriers K
constexpr int BB   = 256;   // batch B
constexpr int TOUT = 10;
constexpr int TIN  = 10;
constexpr float SIG2 = 0.01f;
constexpr float LN2F = 0.69314718055994531f;
constexpr float EPSF = 1e-12f;

typedef float v2f __attribute__((ext_vector_type(2)));
typedef float v8f __attribute__((ext_vector_type(8)));

// Per-k LDS state (one wave32 owns one k). ~11.75 KB each, 4 of them.
struct KState {
  float Hr[MU][NT],   Hi[MU][NT];     // channel (4x64)
  float Fr[NT][NRF],  Fi[NT][NRF];    // analog precoder F (64x8)
  float Wr[NRF][MU],  Wi[NRF][MU];    // digital precoder W (8x4)
  float Vr[NRF][NRF], Vi[NRF][NRF];   // V = W W^H (8x8) / A2 scratch
  float Sr[NRF][NRF], Si[NRF][NRF];   // S = F^H FV (8x8) / term scratch
  float Pr[NT][NRF],  Pi[NT][NRF];    // FV / FW / P scratch (64x8)
  float qr[MU][NRF],  qi[MU][NRF];    // q = conj(H) F (4x8)
  float red[64];                       // small reduction scratch
};

static __device__ __forceinline__ v8f pga_z8() {
  v8f z = {0.f,0.f,0.f,0.f,0.f,0.f,0.f,0.f};
  return z;
}

// D = A*B + C on the fp32 matrix pipe (wave32, 16x16 tile, K-step 4)
static __device__ __forceinline__ v8f pga_wmma(v2f a, v2f b, v8f c) {
  return __builtin_amdgcn_wmma_f32_16x16x4_f32(false, a, false, b, (short)0, c,
                                               false, false);
}

// Complex 16x16 tile GEMM: (cr,ci) += A(16xKd) * B(KdxN<=16), accumulate.
// A fragment: lane m=lane&15, VGPR pair covers K = {ko, ko+1}, ko = 2*(lane>=16).
// B fragment: lane n=lane&15, rows {ka, ka+1}; columns >= nb substituted by 0
// branchlessly (clamped index + cndmask, EXEC stays full for the WMMAs).
// nb must be a power of two (4 or 8) and nb <= ldb.
static __device__ __forceinline__ void pga_cgemm(
    const float* Ar, const float* Ai, int lda,
    const float* Br, const float* Bi, int ldb, int nb,
    int Kd, int lane, v8f& cr, v8f& ci) {
  const int m  = lane & 15;
  const int ko = (lane >> 4) << 1;
  const int n  = lane & 15;
  const bool okn = (n < nb);
  const int nc = n & (nb - 1);          // clamped, always in range
#pragma unroll
  for (int kk = 0; kk < Kd; kk += 4) {
    const int ka = kk + ko;
    v2f a_r = *(const v2f*)(Ar + m * lda + ka);
    v2f a_i = *(const v2f*)(Ai + m * lda + ka);
    float b0 = Br[(ka + 0) * ldb + nc];
    float b1 = Br[(ka + 1) * ldb + nc];
    float c0 = Bi[(ka + 0) * ldb + nc];
    float c1 = Bi[(ka + 1) * ldb + nc];
    v2f b_r, b_i;
    b_r.x = okn ? b0 : 0.f;  b_r.y = okn ? b1 : 0.f;
    b_i.x = okn ? c0 : 0.f;  b_i.y = okn ? c1 : 0.f;
    v2f n_i; n_i.x = -a_i.x; n_i.y = -a_i.y;
    cr = pga_wmma(a_r, b_r, cr);
    cr = pga_wmma(n_i, b_i, cr);
    ci = pga_wmma(a_r, b_i, ci);
    ci = pga_wmma(a_i, b_r, ci);
  }
}

// Complex tile GEMM with B = X^H (X row-major [n][k], conjugated):
// (cr,ci) += A(16xKd) * conj(X)^T. X rows give contiguous v2f B fragments.
static __device__ __forceinline__ void pga_cgemm_bct(
    const float* Ar, const float* Ai, int lda,
    const float* Xr, const float* Xi, int ldx,
    int Kd, int lane, v8f& cr, v8f& ci) {
  const int m  = lane & 15;
  const int ko = (lane >> 4) << 1;
  const int n  = lane & 15;
#pragma unroll
  for (int kk = 0; kk < Kd; kk += 4) {
    const int ka = kk + ko;
    v2f a_r = *(const v2f*)(Ar + m * lda + ka);
    v2f a_i = *(const v2f*)(Ai + m * lda + ka);
    v2f b_r = *(const v2f*)(Xr + n * ldx + ka);
    v2f b_t = *(const v2f*)(Xi + n * ldx + ka);
    v2f b_i; b_i.x = -b_t.x; b_i.y = -b_t.y;
    v2f n_i; n_i.x = -a_i.x; n_i.y = -a_i.y;
    cr = pga_wmma(a_r, b_r, cr);
    cr = pga_wmma(n_i, b_i, cr);
    ci = pga_wmma(a_r, b_i, ci);
    ci = pga_wmma(a_i, b_r, ci);
  }
}

// Store 16x16 f32 C tile (only columns < nb): VGPR v holds row v + 8*(lane>=16).
static __device__ __forceinline__ void pga_store_tile(float* Cr, float* Ci,
                                                      int ldc, int nb, int lane,
                                                      v8f cr, v8f ci) {
  const int n  = lane & 15;
  const int mo = (lane >> 4) << 3;
  if (n < nb) {
#pragma unroll
    for (int v = 0; v < 8; ++v) {
      Cr[(mo + v) * ldc + n] = cr[v];
      Ci[(mo + v) * ldc + n] = ci[v];
    }
  }
}

static __device__ __forceinline__ float pga_wave_sum(float v) {
#pragma unroll
  for (int o = 16; o > 0; o >>= 1) v += __shfl_xor(v, o, 32);
  return v;
}

static __device__ __forceinline__ float pga_san(float x) {
  return __builtin_isfinite(x) ? x : 0.0f;
}

static __device__ __forceinline__ void pga_clampmag(float& dr, float& di) {
  float mag = sqrtf(dr * dr + di * di + 1e-24f);
  float s = fminf(0.5f / (mag + EPSF), 1.0f);
  dr *= s; di *= s;
}

// V = W W^H (8x8)
static __device__ __forceinline__ void pga_compute_V(KState& S, int l) {
#pragma unroll
  for (int e = 0; e < 2; ++e) {
    int idx = l + e * 32; int i = idx >> 3, j = idx & 7;
    float ar = 0.f, ai = 0.f;
#pragma unroll
    for (int m = 0; m < MU; ++m) {
      float xr = S.Wr[i][m], xi = S.Wi[i][m];
      float yr = S.Wr[j][m], yi = S.Wi[j][m];
      ar += xr * yr + xi * yi;
      ai += xi * yr - xr * yi;
    }
    S.Vr[i][j] = ar; S.Vi[i][j] = ai;
  }
}

// q[m][r] = sum_i conj(H[m][i]) F[i][r]
static __device__ __forceinline__ void pga_compute_q(KState& S, int l) {
  int m = l >> 3, r = l & 7;
  float ar = 0.f, ai = 0.f;
  for (int i = 0; i < NT; ++i) {
    float hr = S.Hr[m][i], hi = S.Hi[m][i];
    float fr = S.Fr[i][r], fi = S.Fi[i][r];
    ar += hr * fr + hi * fi;
    ai += hr * fi - hi * fr;
  }
  S.qr[m][r] = ar; S.qi[m][r] = ai;
}

// grad_F_com (per-k contribution) accumulated into shared sums (ds_add_f32)
static __device__ __forceinline__ void pga_grad_F_com(KState& S, int l,
                                                      float* gsum_r,
                                                      float* gsum_i) {
  int m = l >> 3, r = l & 7;
  float qvr[NRF], qvi[NRF];
#pragma unroll
  for (int s2 = 0; s2 < NRF; ++s2) {
    float ar = 0.f, ai = 0.f;
#pragma unroll
    for (int r2 = 0; r2 < NRF; ++r2) {
      float a1 = S.qr[m][r2], b1 = S.qi[m][r2];
      float c1 = S.Vr[r2][s2], d1 = S.Vi[r2][s2];
      ar += a1 * c1 - b1 * d1;
      ai += a1 * d1 + b1 * c1;
    }
    qvr[s2] = ar; qvi[s2] = ai;
  }
  float hr = 0.f, hi = 0.f;
#pragma unroll
  for (int r2 = 0; r2 < NRF; ++r2) {
    float a1 = S.qr[m][r2], b1 = S.qi[m][r2];
    float c1 = S.Wr[r2][m], d1 = S.Wi[r2][m];
    hr += a1 * c1 - b1 * d1;
    hi += a1 * d1 + b1 * c1;
  }
  float tf = 0.f;
#pragma unroll
  for (int s2 = 0; s2 < NRF; ++s2)
    tf += qvr[s2] * S.qr[m][s2] + qvi[s2] * S.qi[m][s2];
  float ti = tf - (hr * hr + hi * hi);
  float d1v = LN2F * (tf + SIG2) + 1e-4f;
  float d2v = LN2F * (ti + SIG2) + 1e-4f;
  float wr = S.Wr[r][m], wi = S.Wi[r][m];
  float qmr = qvr[r] - (hr * wr + hi * wi);   // qV - hFw*conj(W^T)
  float qmi = qvi[r] - (hi * wr - hr * wi);
  S.Sr[m][r] = qvr[r] / d1v - qmr / d2v;      // term (reuse S as scratch)
  S.Si[m][r] = qvi[r] / d1v - qmi / d2v;
  // g[i][r] = sum_m H[m][i] * term[m][r]; accumulate across k with LDS atomics
#pragma unroll
  for (int e = 0; e < 16; ++e) {
    int idx = l + e * 32; int i = idx >> 3, rr = idx & 7;
    float gr = 0.f, gi = 0.f;
#pragma unroll
    for (int mm = 0; mm < MU; ++mm) {
      float h1 = S.Hr[mm][i], h2 = S.Hi[mm][i];
      float t1 = S.Sr[mm][rr], t2 = S.Si[mm][rr];
      gr += h1 * t1 - h2 * t2;
      gi += h1 * t2 + h2 * t1;
    }
    atomicAdd(&gsum_r[idx], gr);   // ds_add_f32
    atomicAdd(&gsum_i[idx], gi);
  }
}

// grad_F_rad (per-k) = 2*(FV*(F^H FV) - R*FV)/Rn2, accumulated into shared sums
static __device__ __forceinline__ void pga_grad_F_rad(
    KState& S, const float* __restrict__ Rr, const float* __restrict__ Ri,
    int l, float rn2k, float* gsum_r, float* gsum_i) {
  // FV = F @ V  -> P
#pragma unroll
  for (int mt = 0; mt < 4; ++mt) {
    v8f cr = pga_z8(), ci = pga_z8();
    pga_cgemm(&S.Fr[mt * 16][0], &S.Fi[mt * 16][0], NRF,
              &S.Vr[0][0], &S.Vi[0][0], NRF, NRF, NRF, l, cr, ci);
    pga_store_tile(&S.Pr[mt * 16][0], &S.Pi[mt * 16][0], NRF, NRF, l, cr, ci);
  }
  // S = F^H FV  (8x8 small)
#pragma unroll
  for (int e = 0; e < 2; ++e) {
    int idx = l + e * 32; int i = idx >> 3, j = idx & 7;
    float ar = 0.f, ai = 0.f;
    for (int t = 0; t < NT; ++t) {
      float fr = S.Fr[t][i], fi = S.Fi[t][i];
      float pr = S.Pr[t][j], pi = S.Pi[t][j];
      ar += fr * pr + fi * pi;
      ai += fr * pi - fi * pr;
    }
    S.Sr[i][j] = ar; S.Si[i][j] = ai;
  }
  __builtin_prefetch(Rr + (l << 6), 0, 0);  // global_prefetch_b8 into L2
  const float sc = 2.0f / rn2k;
  const int n = l & 15, mo = (l >> 4) << 3;
#pragma unroll
  for (int mt = 0; mt < 4; ++mt) {
    v8f tr = pga_z8(), tii = pga_z8();                   // T = R @ FV (Kd=64)
    pga_cgemm(Rr + mt * 16 * NT, Ri + mt * 16 * NT, NT,
              &S.Pr[0][0], &S.Pi[0][0], NRF, NRF, NT, l, tr, tii);
    v8f ur = pga_z8(), ui = pga_z8();                    // U = FV @ S (Kd=8)
    pga_cgemm(&S.Pr[mt * 16][0], &S.Pi[mt * 16][0], NRF,
              &S.Sr[0][0], &S.Si[0][0], NRF, NRF, NRF, l, ur, ui);
    if (n < NRF) {
#pragma unroll
      for (int v = 0; v < 8; ++v) {
        int row = mt * 16 + mo + v;
        atomicAdd(&gsum_r[row * NRF + n], (ur[v] - tr[v]) * sc);
        atomicAdd(&gsum_i[row * NRF + n], (ui[v] - tii[v]) * sc);
      }
    }
  }
}

// ||F@W||_F^2 (wave-reduced, valid in all lanes)
static __device__ __forceinline__ float pga_fwpow(KState& S, int l) {
  float pw = 0.f;
#pragma unroll
  for (int e = 0; e < 8; ++e) {
    int idx = l + e * 32; int t = idx >> 2, m = idx & 3;
    float ar = 0.f, ai = 0.f;
#pragma unroll
    for (int r = 0; r < NRF; ++r) {
      float fr = S.Fr[t][r], fi = S.Fi[t][r];
      float wr = S.Wr[r][m], wi = S.Wi[r][m];
      ar += fr * wr - fi * wi;
      ai += fr * wi + fi * wr;
    }
    pw += ar * ar + ai * ai;
  }
  return pga_wave_sum(pw);
}

// sum-rate for this k (lane0 only)
static __device__ __forceinline__ float pga_rate_k(KState& S) {
  float rate = 0.f;
  for (int m = 0; m < MU; ++m) {
    float qvr[NRF], qvi[NRF];
    for (int s2 = 0; s2 < NRF; ++s2) {
      float ar = 0.f, ai = 0.f;
      for (int r = 0; r < NRF; ++r) {
        float a1 = S.qr[m][r], b1 = S.qi[m][r];
        float c1 = S.Vr[r][s2], d1 = S.Vi[r][s2];
        ar += a1 * c1 - b1 * d1;
        ai += a1 * d1 + b1 * c1;
      }
      qvr[s2] = ar; qvi[s2] = ai;
    }
    float hr = 0.f, hi = 0.f;
    for (int r = 0; r < NRF; ++r) {
      hr += S.qr[m][r] * S.Wr[r][m] - S.qi[m][r] * S.Wi[r][m];
      hi += S.qr[m][r] * S.Wi[r][m] + S.qi[m][r] * S.Wr[r][m];
    }
    float tf = 0.f;
    for (int s2 = 0; s2 < NRF; ++s2)
      tf += qvr[s2] * S.qr[m][s2] + qvi[s2] * S.qi[m][s2];
    float ti = tf - (hr * hr + hi * hi);
    rate += log2f(tf + SIG2) - log2f(ti + SIG2);
  }
  return rate;
}

// tau_k = ||F V F^H - R||_F^2 / Rn2   (uses V already computed; WMMA tiles)
static __device__ __forceinline__ float pga_tau_k(KState& S,
                                                  const float* __restrict__ Rr,
                                                  const float* __restrict__ Ri,
                                                  int l, float rn2k) {
#pragma unroll
  for (int mt = 0; mt < 4; ++mt) {                       // P = F @ V
    v8f cr = pga_z8(), ci = pga_z8();
    pga_cgemm(&S.Fr[mt * 16][0], &S.Fi[mt * 16][0], NRF,
              &S.Vr[0][0], &S.Vi[0][0], NRF, NRF, NRF, l, cr, ci);
    pga_store_tile(&S.Pr[mt * 16][0], &S.Pi[mt * 16][0], NRF, NRF, l, cr, ci);
  }
  float acc = 0.f;
  const int n = l & 15, mo = (l >> 4) << 3;
  for (int mt = 0; mt < 4; ++mt)
    for (int nt = 0; nt < 4; ++nt) {                     // E tile = P F^H - R
      v8f cr = pga_z8(), ci = pga_z8();
      pga_cgemm_bct(&S.Pr[mt * 16][0], &S.Pi[mt * 16][0], NRF,
                    &S.Fr[nt * 16][0], &S.Fi[nt * 16][0], NRF, NRF, l, cr, ci);
#pragma unroll
      for (int v = 0; v < 8; ++v) {
        int row = mt * 16 + mo + v, col = nt * 16 + n;
        float rr = Rr[row * NT + col], ri = Ri[row * NT + col];
        float er = cr[v] - rr, ei = ci[v] - ri;
        acc += er * er + ei * ei;
      }
    }
  return pga_wave_sum(acc) / rn2k;
}

extern "C" __global__ __launch_bounds__(128, 1)
void PGA_Unfold_J10_PC_kernel(const float* __restrict__ Hr_g,
                              const float* __restrict__ Hi_g,
                              const float* __restrict__ Rr_g,
                              const float* __restrict__ Ri_g,
                              const float* __restrict__ W0r_g,
                              const float* __restrict__ W0i_g,
                              const float* __restrict__ Ph_g,
                              const float* __restrict__ St_g,
                              const float* __restrict__ Pt_g,
                              float* __restrict__ out) {
  __shared__ KState ksAll[KC];                 // ~47 KB
  __shared__ float GS[4][NT * NRF];            // 8 KB: k-summed gc/gr, T2 scratch
  __shared__ float rn2s[KC], rateK[KC], tauK[KC];

  const int b   = blockIdx.x;
  const int tid = threadIdx.x;
  const int w   = tid >> 5;   // wave = k
  const int l   = tid & 31;
  KState& S = ksAll[w];

  __builtin_amdgcn_s_wait_tensorcnt(0);

  const float Pt = Pt_g[0];
  const size_t kb = (size_t)(w * BB + b);
  const float* __restrict__ Rr = Rr_g + kb * (size_t)(NT * NT);
  const float* __restrict__ Ri = Ri_g + kb * (size_t)(NT * NT);

  // ---------------- init ----------------
#pragma unroll
  for (int e = 0; e < 8; ++e) {                 // H
    int idx = l + e * 32;
    S.Hr[idx >> 6][idx & 63] = Hr_g[kb * (size_t)(MU * NT) + idx];
    S.Hi[idx >> 6][idx & 63] = Hi_g[kb * (size_t)(MU * NT) + idx];
  }
#pragma unroll
  for (int e = 0; e < 16; ++e) {                // F = exp(i*phase) * mask
    int idx = l + e * 32; int t = idx >> 3, r = idx & 7;
    float ph = Ph_g[(size_t)b * (NT * NRF) + idx];
    float sn, cs; __sincosf(ph, &sn, &cs);
    bool msk = ((t >> 3) == r);
    S.Fr[t][r] = msk ? cs : 0.f;
    S.Fi[t][r] = msk ? sn : 0.f;
  }
  {                                             // W0
    int a = l >> 2, m = l & 3;
    S.Wr[a][m] = W0r_g[kb * (size_t)(NRF * MU) + l];
    S.Wi[a][m] = W0i_g[kb * (size_t)(NRF * MU) + l];
  }
  {                                             // Rn2[k] from R[k, b=0]
    const float* R0r = Rr_g + (size_t)(w * BB) * (NT * NT);
    const float* R0i = Ri_g + (size_t)(w * BB) * (NT * NT);
    float acc = 0.f;
    for (int e = 0; e < 128; ++e) {
      int idx = l + e * 32;
      float xr = R0r[idx], xi = R0i[idx];
      acc += xr * xr + xi * xi;
    }
    acc = pga_wave_sum(acc);
    if (l == 0) rn2s[w] = acc;
  }
  {                                             // W = normalize_W(F, W0)
    float pw = pga_fwpow(S, l);
    float scw = pga_san(sqrtf(Pt / (pw + EPSF)));
    int a = l >> 2, m = l & 3;
    S.Wr[a][m] *= scw;
    S.Wi[a][m] *= scw;
  }
  __syncthreads();
  const float rn2k = rn2s[w];

  // ----- initial rate / tau (column 0) -----
  pga_compute_V(S, l);
  pga_compute_q(S, l);
  if (l == 0) rateK[w] = pga_rate_k(S);
  {
    float tau = pga_tau_k(S, Rr, Ri, l, rn2k);
    if (l == 0) tauK[w] = tau;
  }
  __syncthreads();
  if (tid == 0) {
    out[(size_t)b * (TOUT + 1)] =
        0.25f * (rateK[0] + rateK[1] + rateK[2] + rateK[3]);
    out[(size_t)BB * (TOUT + 1) + (size_t)b * (TOUT + 1)] =
        0.25f * (tauK[0] + tauK[1] + tauK[2] + tauK[3]);
  }
  __syncthreads();

  // ---------------- unfolded PGA ----------------
  for (int to = 0; to < TOUT; ++to) {
    pga_compute_V(S, l);  // W fixed across the inner loop

    for (int ji = 0; ji < TIN; ++ji) {
      // zero the cross-k gradient accumulators
#pragma unroll
      for (int e = 0; e < 16; ++e) GS[tid >> 5][((tid & 31) + e * 32)] = 0.f;
      __syncthreads();

      const float sstep = St_g[(ji * 10 + to) * (KC + 1)];
      pga_compute_q(S, l);
      pga_grad_F_com(S, l, &GS[0][0], &GS[1][0]);
      pga_grad_F_rad(S, Rr, Ri, l, rn2k, &GS[2][0], &GS[3][0]);
      __syncthreads();

      // F update with k-averaged, magnitude-clamped gradients
#pragma unroll
      for (int e = 0; e < 16; ++e) {
        int idx = l + e * 32; int t = idx >> 3, r = idx & 7;
        float dcr = sstep * 0.25f * GS[0][idx];
        float dci = sstep * 0.25f * GS[1][idx];
        float drr = sstep * 0.25f * GS[2][idx];
        float dri = sstep * 0.25f * GS[3][idx];
        pga_clampmag(dcr, dci);
        pga_clampmag(drr, dri);
        bool msk = ((t >> 3) == r);
        float fr = S.Fr[t][r] + dcr - drr;
        float fi = S.Fi[t][r] + dci - dri;
        S.Fr[t][r] = msk ? pga_san(fr) : 0.f;
        S.Fi[t][r] = msk ? pga_san(fi) : 0.f;
      }
      __syncthreads();
      {  // normalize_F (per-k power)
        float pw = pga_fwpow(S, l);
        float scf = pga_san(sqrtf(Pt / (pw + EPSF)));
#pragma unroll
        for (int e = 0; e < 16; ++e) {
          int idx = l + e * 32; int t = idx >> 3, r = idx & 7;
          S.Fr[t][r] = pga_san(S.Fr[t][r] * scf);
          S.Fi[t][r] = pga_san(S.Fi[t][r] * scf);
        }
      }
      __syncthreads();
    }

    // project_unit(F, mask)
#pragma unroll
    for (int e = 0; e < 16; ++e) {
      int idx = l + e * 32; int t = idx >> 3, r = idx & 7;
      bool msk = ((t >> 3) == r);
      float fr = S.Fr[t][r], fi = S.Fi[t][r];
      float mag = sqrtf(fr * fr + fi * fi + 1e-24f);
      float pr2, pi2;
      if (mag > EPSF) { pr2 = fr / mag; pi2 = fi / mag; }
      else            { pr2 = 1.f;      pi2 = 0.f; }
      S.Fr[t][r] = msk ? pga_san(pr2) : 0.f;
      S.Fi[t][r] = msk ? pga_san(pi2) : 0.f;
    }

    // ---------------- W step ----------------
    pga_compute_q(S, l);
    const int a = l >> 2, m = l & 3;
    float gwc_r, gwc_i;
    {  // grad_W_com: u = conj(W)^T conj(q); g = conj(q)*conj(u_mm)/d /K
      float ur_[MU], ui_[MU];
#pragma unroll
      for (int j = 0; j < MU; ++j) {
        float sr = 0.f, si = 0.f;
#pragma unroll
        for (int aa = 0; aa < NRF; ++aa) {
          float wr = S.Wr[aa][j], wi = S.Wi[aa][j];
          float q1 = S.qr[m][aa], q2 = S.qi[m][aa];
          sr += wr * q1 - wi * q2;
          si += -(wr * q2 + wi * q1);
        }
        ur_[j] = sr; ui_[j] = si;
      }
      float dm = 0.f;
#pragma unroll
      for (int j = 0; j < MU; ++j) dm += ur_[j] * ur_[j] + ui_[j] * ui_[j];
      dm = LN2F * (dm + SIG2);
      float cr_ = ur_[m] / dm, ci_ = -ui_[m] / dm;
      float rr_ = S.qr[m][a], ri_ = -S.qi[m][a];
      gwc_r = (rr_ * cr_ - ri_ * ci_) * (1.0f / KC);
      gwc_i = (rr_ * ci_ + ri_ * cr_) * (1.0f / KC);
    }
    float gwr_r, gwr_i;
    {  // grad_W_rad = 2((F^H FW)(FW^H FW) - F^H (R FW))/Rn2/K
#pragma unroll
      for (int mt = 0; mt < 4; ++mt) {               // FW = F @ W -> P
        v8f cr = pga_z8(), ci = pga_z8();
        pga_cgemm(&S.Fr[mt * 16][0], &S.Fi[mt * 16][0], NRF,
                  &S.Wr[0][0], &S.Wi[0][0], MU, MU, NRF, l, cr, ci);
        pga_store_tile(&S.Pr[mt * 16][0], &S.Pi[mt * 16][0], NRF, MU, l, cr, ci);
      }
      if (l < 16) {                                  // B2 = FW^H FW (4x4)
        int i = l >> 2, j = l & 3;
        float ar = 0.f, ai = 0.f;
        for (int t = 0; t < NT; ++t) {
          float x1 = S.Pr[t][i], x2 = S.Pi[t][i];
          float y1 = S.Pr[t][j], y2 = S.Pi[t][j];
          ar += x1 * y1 + x2 * y2;
          ai += x1 * y2 - x2 * y1;
        }
        S.red[2 * l] = ar; S.red[2 * l + 1] = ai;
      }
      {                                              // A2 = F^H FW (8x4)
        float ar = 0.f, ai = 0.f;
        for (int t = 0; t < NT; ++t) {
          float fr = S.Fr[t][a], fi = S.Fi[t][a];
          float y1 = S.Pr[t][m], y2 = S.Pi[t][m];
          ar += fr * y1 + fi * y2;
          ai += fr * y2 - fi * y1;
        }
        S.Vr[a][m] = ar; S.Vi[a][m] = ai;            // V scratch (rebuilt later)
      }
      // T2 = R @ FW (Kd=64), stored in this wave's quarter of GS
      float* T2r = &GS[w][0];                        // [64][4]
      float* T2i = &GS[w][NT * MU];                  // [64][4]
      const int n = l & 15, mo = (l >> 4) << 3;
#pragma unroll
      for (int mt = 0; mt < 4; ++mt) {
        v8f tr = pga_z8(), tii = pga_z8();
        pga_cgemm(Rr + mt * 16 * NT, Ri + mt * 16 * NT, NT,
                  &S.Pr[0][0], &S.Pi[0][0], NRF, MU, NT, l, tr, tii);
        if (n < MU) {
#pragma unroll
          for (int v = 0; v < 8; ++v) {
            int row = mt * 16 + mo + v;
            T2r[row * MU + n] = tr[v];
            T2i[row * MU + n] = tii[v];
          }
        }
      }
      float c2r = 0.f, c2i = 0.f;                    // C2 = F^H T2 (8x4)
      for (int t = 0; t < NT; ++t) {
        float fr = S.Fr[t][a], fi = S.Fi[t][a];
        float y1 = T2r[t * MU + m], y2 = T2i[t * MU + m];
        c2r += fr * y1 + fi * y2;
        c2i += fr * y2 - fi * y1;
      }
      float abr = 0.f, abi = 0.f;                    // (A2 @ B2)[a][m]
#pragma unroll
      for (int x = 0; x < MU; ++x) {
        float a1 = S.Vr[a][x], a2 = S.Vi[a][x];
        float b1 = S.red[2 * (x * MU + m)], b2 = S.red[2 * (x * MU + m) + 1];
        abr += a1 * b1 - a2 * b2;
        abi += a1 * b2 + a2 * b1;
      }
      gwr_r = 2.f * (abr - c2r) / rn2k * (1.0f / KC);
      gwr_i = 2.f * (abi - c2i) / rn2k * (1.0f / KC);
    }
    {  // W update + normalize_W
      const float sk = St_g[to * (KC + 1) + 1 + w];
      float dcr = sk * gwc_r, dci = sk * gwc_i; pga_clampmag(dcr, dci);
      float drr = sk * gwr_r, dri = sk * gwr_i; pga_clampmag(drr, dri);
      S.Wr[a][m] = pga_san(S.Wr[a][m] + dcr - drr);
      S.Wi[a][m] = pga_san(S.Wi[a][m] + dci - dri);
      float pw = pga_fwpow(S, l);
      float scw = pga_san(sqrtf(Pt / (pw + EPSF)));
      S.Wr[a][m] = pga_san(S.Wr[a][m] * scw);
      S.Wi[a][m] = pga_san(S.Wi[a][m] * scw);
    }

    // ----- rate / tau with updated F, W -----
    pga_compute_V(S, l);
    if (l == 0) rateK[w] = pga_rate_k(S);
    {
      float tau = pga_tau_k(S, Rr, Ri, l, rn2k);
      if (l == 0) tauK[w] = tau;
    }
    __syncthreads();
    if (tid == 0) {
      out[(size_t)b * (TOUT + 1) + (to + 1)] =
          0.25f * (rateK[0] + rateK[1] + rateK[2] + rateK[3]);
      out[(size_t)BB * (TOUT + 1) + (size_t)b * (TOUT + 1) + (to + 1)] =
          0.25f * (tauK[0] + tauK[1] + tauK[2] + tauK[3]);
    }
    __syncthreads();
  }

  // ---------------- write F, W (complex64 interleaved) ----------------
  const size_t oF = (size_t)2 * BB * (TOUT + 1);
  const size_t oW = oF + (size_t)2 * KC * BB * NT * NRF;
#pragma unroll
  for (int e = 0; e < 16; ++e) {
    int idx = l + e * 32; int t = idx >> 3, r = idx & 7;
    size_t p = oF + (kb * (size_t)(NT * NRF) + idx) * 2;
    out[p]     = S.Fr[t][r];
    out[p + 1] = S.Fi[t][r];
  }
  {
    int a2 = l >> 2, m2 = l & 3;
    size_t p = oW + (kb * (size_t)(NRF * MU) + l) * 2;
    out[p]     = S.Wr[a2][m2];
    out[p + 1] = S.Wi[a2][m2];
  }
}

extern "C" void kernel_launch(void* const* d_in, const int* in_sizes, int n_in,
                              void* d_out, int out_size, void* d_ws,
                              size_t ws_size, hipStream_t stream) {
  (void)in_sizes; (void)n_in; (void)d_ws; (void)ws_size; (void)out_size;
  const float* Hr  = (const float*)d_in[0];
  const float* Hi  = (const float*)d_in[1];
  const float* Rr  = (const float*)d_in[2];
  const float* Ri  = (const float*)d_in[3];
  const float* W0r = (const float*)d_in[4];
  const float* W0i = (const float*)d_in[5];
  const float* Ph  = (const float*)d_in[6];
  const float* St  = (const float*)d_in[7];
  const float* Pt  = (const float*)d_in[8];
  PGA_Unfold_J10_PC_kernel<<<dim3(BB), dim3(128), 0, stream>>>(
      Hr, Hi, Rr, Ri, W0r, W0i, Ph, St, Pt, (float*)d_out);
}